// HTGNN_46145128628374
// MI455X (gfx1250) — compile-verified
//
#include <hip/hip_runtime.h>
#include <math.h>

// ---------------------------------------------------------------------------
// HTGNN forward (HAN conv x2 + temporal MHA + gated MLP head) for gfx1250.
// All dense GEMMs use V_WMMA_F32_16X16X4_F32 (fp32 in/out, matches reference
// precision; workload is HBM-bound so f32 matrix ops are free).
// ---------------------------------------------------------------------------

#define HIDN  128
#define NHEAD 4
#define HDIM  32
#define SEQN  10
#define NTXN  50000
#define NUSR  20000
#define NMER  10000
#define NEDGE 200000
#define TCH   5000            // temporal chunk (txn nodes)
#define TCHROWS (TCH * SEQN)  // 50000 rows per chunk

typedef __attribute__((ext_vector_type(2))) float v2f;
typedef __attribute__((ext_vector_type(8))) float v8f;

// ---------------------------------------------------------------------------
// WMMA f32 16x16x4 tile mainloop: acc(16x16) += A[16xK] * W[KxN] tile.
// A row-major lda=K, W row-major ldw=N. Requires mBase+16<=M (all our M are
// multiples of 16). A-frag: lanes 0-15 -> K+{0,1}, lanes 16-31 -> K+{2,3}.
// B-frag mirrors (rows K+0/K+1 lanes 0-15, K+2/K+3 lanes 16-31).
// C/D: VGPR r holds (M=r, lanes0-15) and (M=r+8, lanes16-31).
// ---------------------------------------------------------------------------
__device__ __forceinline__ v8f wmma_tile_f32(const float* __restrict__ A,
                                             const float* __restrict__ W,
                                             int K, int ldw,
                                             int mBase, int nBase, int lane) {
  const int half = lane >> 4;
  const int l16  = lane & 15;
  const float* __restrict__ Ar = A + (size_t)(mBase + l16) * K + 2 * half;
  const float* __restrict__ Wp = W + (size_t)(2 * half) * ldw + nBase + l16;
  v8f acc = {};
  for (int k = 0; k < K; k += 4) {
    v2f a, b;
    a[0] = Ar[k];
    a[1] = Ar[k + 1];
    b[0] = Wp[(size_t)k * ldw];
    b[1] = Wp[(size_t)(k + 1) * ldw];
    acc = __builtin_amdgcn_wmma_f32_16x16x4_f32(false, a, false, b,
                                                (short)0, acc, false, false);
  }
  return acc;
}

// act: 0 = none, 1 = relu, 2 = sigmoid
__global__ void gemm_bias_act_k(const float* __restrict__ A, const float* __restrict__ W,
                                const float* __restrict__ bias, float* __restrict__ C,
                                int M, int K, int N, int act) {
  const int lane = threadIdx.x & 31, wave = threadIdx.x >> 5;
  const int mBase = blockIdx.x * 16;
  const int nBase = (blockIdx.y * 8 + wave) * 16;
  if (nBase >= N) return;                      // wave-uniform exit (EXEC stays full for WMMA)
  v8f acc = wmma_tile_f32(A, W, K, N, mBase, nBase, lane);
  const int half = lane >> 4;
  const int col  = nBase + (lane & 15);
  const float bb = bias ? bias[col] : 0.f;
#pragma unroll
  for (int r = 0; r < 8; ++r) {
    const int row = mBase + r + 8 * half;
    float v = acc[r] + bb;
    if (act == 1)      v = fmaxf(v, 0.f);
    else if (act == 2) v = 1.f / (1.f + __expf(-v));
    C[(size_t)row * N + col] = v;
  }
}

// input projection + eval-mode BatchNorm (mean=0, var=1) + relu
__global__ void gemm_projbn_k(const float* __restrict__ A, const float* __restrict__ W,
                              const float* __restrict__ bias, const float* __restrict__ g,
                              const float* __restrict__ beta, float* __restrict__ C,
                              int M, int K, int N) {
  const int lane = threadIdx.x & 31, wave = threadIdx.x >> 5;
  const int mBase = blockIdx.x * 16;
  const int nBase = wave * 16;   // N == 128
  v8f acc = wmma_tile_f32(A, W, K, N, mBase, nBase, lane);
  const int half = lane >> 4;
  const int col  = nBase + (lane & 15);
  const float bb = bias[col];
  const float gs = g[col] * rsqrtf(1.0f + 1e-5f);
  const float be = beta[col];
#pragma unroll
  for (int r = 0; r < 8; ++r) {
    const int row = mBase + r + 8 * half;
    float v = (acc[r] + bb) * gs + be;
    C[(size_t)row * N + col] = fmaxf(v, 0.f);
  }
}

// seq = txn_seq @ tproj_w + tproj_b + [sin|cos](dt*time_w + time_b)
__global__ void gemm_tproj_k(const float* __restrict__ A, const float* __restrict__ W,
                             const float* __restrict__ bias, const float* __restrict__ dt,
                             const float* __restrict__ tw, const float* __restrict__ tb,
                             float* __restrict__ C, int M, int K, int N) {
  const int lane = threadIdx.x & 31, wave = threadIdx.x >> 5;
  const int mBase = blockIdx.x * 16;
  const int nBase = wave * 16;   // N == 128
  v8f acc = wmma_tile_f32(A, W, K, N, mBase, nBase, lane);
  const int half = lane >> 4;
  const int col  = nBase + (lane & 15);
  const float bb = bias[col];
  const int   fi = (col < 64) ? col : (col - 64);
  const float fw = tw[fi], fb = tb[fi];
#pragma unroll
  for (int r = 0; r < 8; ++r) {
    const int row = mBase + r + 8 * half;
    const float t = dt[row];
    const float ph = t * fw + fb;
    const float enc = (col < 64) ? sinf(ph) : cosf(ph);
    C[(size_t)row * N + col] = acc[r] + bb + enc;
  }
}

// semantic-attention score: score += (1/M) * sum_n sum_j tanh((S@kw)[n,j]+kb[j])*q[j]
__global__ void semantic_score_k(const float* __restrict__ S, const float* __restrict__ kw,
                                 const float* __restrict__ kb, const float* __restrict__ q,
                                 float* __restrict__ score, float invN) {
  const int lane = threadIdx.x & 31, wave = threadIdx.x >> 5;
  const int mBase = blockIdx.x * 16;
  const int nBase = wave * 16;   // N == 128
  v8f acc = wmma_tile_f32(S, kw, HIDN, HIDN, mBase, nBase, lane);
  const int col = nBase + (lane & 15);
  const float qq = q[col], kbb = kb[col];
  float p = 0.f;
#pragma unroll
  for (int r = 0; r < 8; ++r) p += tanhf(acc[r] + kbb);
  atomicAdd(score, p * qq * invN);
}

// ---------------------------------------------------------------------------
// Elementwise / graph kernels
// ---------------------------------------------------------------------------
__global__ void fillf_k(float* __restrict__ p, float v, int n) {
  int i = blockIdx.x * blockDim.x + threadIdx.x;
  if (i < n) p[i] = v;
}
__global__ void relu_k(float* __restrict__ p, int n) {
  int i = blockIdx.x * blockDim.x + threadIdx.x;
  if (i < n) p[i] = fmaxf(p[i], 0.f);
}
__global__ void add_k(const float* __restrict__ a, const float* __restrict__ b,
                      float* __restrict__ c, int n) {
  int i = blockIdx.x * blockDim.x + threadIdx.x;
  if (i < n) c[i] = a[i] + b[i];
}
__global__ void combine2_k(const float* __restrict__ o0, const float* __restrict__ o1,
                           const float* __restrict__ w, float* __restrict__ res, int n) {
  int i = blockIdx.x * blockDim.x + threadIdx.x;
  if (i < n) res[i] = w[0] * o0[i] + w[1] * o1[i];
}
__global__ void softmax2_k(const float* __restrict__ s, float* __restrict__ w) {
  if (threadIdx.x == 0 && blockIdx.x == 0) {
    float m = fmaxf(s[0], s[1]);
    float e0 = __expf(s[0] - m), e1 = __expf(s[1] - m);
    float inv = 1.f / (e0 + e1);
    w[0] = e0 * inv; w[1] = e1 * inv;
  }
}
__global__ void build_cat_k(const float* __restrict__ a, const float* __restrict__ b,
                            float* __restrict__ cat, int n /* = N*128 */) {
  int i = blockIdx.x * blockDim.x + threadIdx.x;
  if (i >= n) return;
  int row = i >> 7, c = i & 127;
  cat[(size_t)row * 256 + c]       = a[i];
  cat[(size_t)row * 256 + 128 + c] = b[i];
}
__global__ void apply_gate_k(float* __restrict__ cat, const float* __restrict__ g, int n) {
  int i = blockIdx.x * blockDim.x + threadIdx.x;
  if (i >= n) return;
  int row = i >> 7, c = i & 127;
  cat[(size_t)row * 256 + 128 + c] *= g[i];
}

// a_{n,h} = sum_j xp[n, h*32+j] * att[h*32+j]
__global__ void att_score_k(const float* __restrict__ xp, const float* __restrict__ att,
                            float* __restrict__ out, int n4) {
  int i = blockIdx.x * blockDim.x + threadIdx.x;
  if (i >= n4) return;
  int n = i >> 2, h = i & 3;
  const float* xr = xp + (size_t)n * HIDN + h * HDIM;
  const float* ar = att + h * HDIM;
  float s = 0.f;
#pragma unroll
  for (int j = 0; j < HDIM; ++j) s += xr[j] * ar[j];
  out[i] = s;
}

__device__ __forceinline__ void atomicMaxF32(float* addr, float val) {
  unsigned* ua = (unsigned*)addr;
  unsigned old = *ua;
  while (__uint_as_float(old) < val) {
    unsigned assumed = old;
    old = atomicCAS(ua, assumed, __float_as_uint(val));
    if (old == assumed) break;
  }
}

// pass 1: per-(edge,head) leaky-relu logit + segment max over dst
__global__ void edge_logits_max_k(const int* __restrict__ ei, const float* __restrict__ as,
                                  const float* __restrict__ ad, float* __restrict__ aE,
                                  float* __restrict__ m, int E) {
  int i = blockIdx.x * blockDim.x + threadIdx.x;
  if (i >= E * 4) return;
  int e = i >> 2, h = i & 3;
  int s = ei[e], d = ei[E + e];
  float a = as[s * 4 + h] + ad[d * 4 + h];
  a = (a < 0.f) ? 0.2f * a : a;
  aE[i] = a;
  atomicMaxF32(&m[d * 4 + h], a);
}
// pass 2: exp(a - max) and segment sum
__global__ void edge_exp_sum_k(const int* __restrict__ ei, float* __restrict__ aE,
                               const float* __restrict__ m, float* __restrict__ ss, int E) {
  int i = blockIdx.x * blockDim.x + threadIdx.x;
  if (i >= E * 4) return;
  int e = i >> 2, h = i & 3;
  int d = ei[E + e];
  float ex = __expf(aE[i] - m[d * 4 + h]);
  aE[i] = ex;
  atomicAdd(&ss[d * 4 + h], ex);
}
// pass 3: alpha = exp / (sum + eps)
__global__ void edge_norm_k(const int* __restrict__ ei, float* __restrict__ aE,
                            const float* __restrict__ ss, int E) {
  int i = blockIdx.x * blockDim.x + threadIdx.x;
  if (i >= E * 4) return;
  int e = i >> 2, h = i & 3;
  int d = ei[E + e];
  aE[i] = aE[i] / (ss[d * 4 + h] + 1e-16f);
}
// pass 4: out[d, c] += xp_src[s, c] * alpha[e, c/32]   (one thread per edge-channel)
__global__ void edge_scatter_k(const int* __restrict__ ei, const float* __restrict__ xp,
                               const float* __restrict__ aE, float* __restrict__ out, int E) {
  int i = blockIdx.x * blockDim.x + threadIdx.x;
  if (i >= E * HIDN) return;
  int e = i >> 7, c = i & 127, h = c >> 5;
  int s = ei[e], d = ei[E + e];
  atomicAdd(&out[(size_t)d * HIDN + c], xp[(size_t)s * HIDN + c] * aE[e * 4 + h]);
}

// LayerNorm over 128 channels, one wave per row (optionally relu the input)
__global__ void layernorm128_k(const float* __restrict__ x, const float* __restrict__ g,
                               const float* __restrict__ b, float* __restrict__ y,
                               int n, int relu_in) {
  const int lane = threadIdx.x & 31, wave = threadIdx.x >> 5;
  const int row = blockIdx.x * 8 + wave;
  if (row >= n) return;
  const float* xr = x + (size_t)row * HIDN + lane * 4;
  float v[4];
#pragma unroll
  for (int i = 0; i < 4; ++i) {
    v[i] = xr[i];
    if (relu_in) v[i] = fmaxf(v[i], 0.f);
  }
  float s = v[0] + v[1] + v[2] + v[3];
#pragma unroll
  for (int m = 1; m < 32; m <<= 1) s += __shfl_xor(s, m, 32);
  const float mu = s * (1.f / 128.f);
  float q = 0.f;
#pragma unroll
  for (int i = 0; i < 4; ++i) { float d = v[i] - mu; q += d * d; }
#pragma unroll
  for (int m = 1; m < 32; m <<= 1) q += __shfl_xor(q, m, 32);
  const float inv = rsqrtf(q * (1.f / 128.f) + 1e-5f);
  float* yr = y + (size_t)row * HIDN + lane * 4;
  const int c = lane * 4;
#pragma unroll
  for (int i = 0; i < 4; ++i) yr[i] = g[c + i] * (v[i] - mu) * inv + b[c + i];
}

// Per-txn MHA over SEQ=10: one wave per node; lane = head*8 + part, 4 ch each.
__global__ void mha10_k(const float* __restrict__ q, const float* __restrict__ kh,
                        const float* __restrict__ vh, float* __restrict__ ctx, int n) {
  const int lane = threadIdx.x & 31, wave = threadIdx.x >> 5;
  const int node = blockIdx.x * 8 + wave;
  if (node >= n) return;
  const int c = (lane >> 3) * HDIM + (lane & 7) * 4;
  const float* qp = q + (size_t)node * HIDN + c;
  const float q0 = qp[0], q1 = qp[1], q2 = qp[2], q3 = qp[3];
  float sc[SEQN];
#pragma unroll
  for (int s = 0; s < SEQN; ++s) {
    const float* kp = kh + ((size_t)node * SEQN + s) * HIDN + c;
    float d = q0 * kp[0] + q1 * kp[1] + q2 * kp[2] + q3 * kp[3];
    d += __shfl_xor(d, 1, 32);
    d += __shfl_xor(d, 2, 32);
    d += __shfl_xor(d, 4, 32);
    sc[s] = d * 0.17677669529663687f;   // 1/sqrt(32)
  }
  float mx = sc[0];
#pragma unroll
  for (int s = 1; s < SEQN; ++s) mx = fmaxf(mx, sc[s]);
  float sum = 0.f;
#pragma unroll
  for (int s = 0; s < SEQN; ++s) { sc[s] = __expf(sc[s] - mx); sum += sc[s]; }
  const float inv = 1.f / sum;
  float a0 = 0.f, a1 = 0.f, a2 = 0.f, a3 = 0.f;
#pragma unroll
  for (int s = 0; s < SEQN; ++s) {
    const float w = sc[s] * inv;
    const float* vp = vh + ((size_t)node * SEQN + s) * HIDN + c;
    a0 += w * vp[0]; a1 += w * vp[1]; a2 += w * vp[2]; a3 += w * vp[3];
  }
  float* op = ctx + (size_t)node * HIDN + c;
  op[0] = a0; op[1] = a1; op[2] = a2; op[3] = a3;
}

// final head: out[n] = z2[n,:64] . h3_w + h3_b  (one wave per row)
__global__ void head_out_k(const float* __restrict__ z, const float* __restrict__ w,
                           const float* __restrict__ b, float* __restrict__ out, int n) {
  const int lane = threadIdx.x & 31, wave = threadIdx.x >> 5;
  const int row = blockIdx.x * 8 + wave;
  if (row >= n) return;
  const float* zr = z + (size_t)row * 64;
  float p = zr[lane] * w[lane] + zr[lane + 32] * w[lane + 32];
#pragma unroll
  for (int m = 1; m < 32; m <<= 1) p += __shfl_xor(p, m, 32);
  if (lane == 0) out[row] = p + b[0];
}

// ---------------------------------------------------------------------------
// Host orchestration
// ---------------------------------------------------------------------------
// Input order: setup_inputs() dict order; 'params' flattened as a jax pytree
// (dict keys sorted recursively) => 79 flat device pointers.
enum {
  IN_XTXN = 0, IN_XUSR, IN_XMER,
  IN_EI_UT, IN_EI_TU, IN_EI_MT, IN_EI_TM,
  IN_SEQ, IN_DT,
  IN_BK, IN_BO, IN_BQ, IN_BV,          // 9..12
  C1BASE = 13,                          // conv1: 17 leaves (see offsets below)
  C2BASE = 30,                          // conv2: 17 leaves
  IN_GATE_B = 47, IN_GATE_W, IN_H1B, IN_H1W, IN_H2B, IN_H2W, IN_H3B, IN_H3W,
  IN_LN1B = 55, IN_LN1G, IN_LN2B, IN_LN2G,
  PRJBASE = 59,                         // proj: merchant{b,beta,g,w} txn{..} user{..}
  IN_TIMEB = 71, IN_TIMEW, IN_TPROJB, IN_TPROJW,
  IN_WK = 75, IN_WO, IN_WQ, IN_WV
};
// conv block relative offsets (sorted keys):
//  +0..3  att_dst: pays, receives, rev_pays, rev_receives
//  +4..7  att_src: pays, receives, rev_pays, rev_receives
//  +8 k_b   +9 k_w
//  +10 proj.merchant.b  +11 .w   +12 proj.txn.b  +13 .w   +14 proj.user.b  +15 .w
//  +16 q

struct ConvP {
  const float* adst[4];
  const float* asrc[4];
  const float *kw, *kb, *q;
  const float *pwT, *pbT, *pwU, *pbU, *pwM, *pbM;
};

extern "C" void kernel_launch(void* const* d_in, const int* in_sizes, int n_in,
                              void* d_out, int out_size, void* d_ws, size_t ws_size,
                              hipStream_t stream) {
  (void)in_sizes; (void)n_in; (void)out_size; (void)ws_size;
  auto PF = [&](int i) { return (const float*)d_in[i]; };
  auto PI = [&](int i) { return (const int*)d_in[i]; };
  auto cdiv = [](long a, long b) { return (int)((a + b - 1) / b); };

  // ---- workspace layout (floats, 256B aligned blocks); total ~500 MB ----
  char* wsb = (char*)d_ws;
  size_t off = 0;
  auto alloc = [&](size_t nf) {
    float* p = (float*)(wsb + off);
    off = (off + nf * sizeof(float) + 255) & ~(size_t)255;
    return p;
  };
  float* h0t = alloc((size_t)NTXN * HIDN);
  float* h0u = alloc((size_t)NUSR * HIDN);
  float* h0m = alloc((size_t)NMER * HIDN);
  float* xpt = alloc((size_t)NTXN * HIDN);
  float* xpu = alloc((size_t)NUSR * HIDN);
  float* xpm = alloc((size_t)NMER * HIDN);
  float* h1t = alloc((size_t)NTXN * HIDN);
  float* h1u = alloc((size_t)NUSR * HIDN);
  float* h1m = alloc((size_t)NMER * HIDN);
  float* ot0 = alloc((size_t)NTXN * HIDN);
  float* ot1 = alloc((size_t)NTXN * HIDN);
  float* ou  = alloc((size_t)NUSR * HIDN);
  float* om  = alloc((size_t)NMER * HIDN);
  float* rest = alloc((size_t)NTXN * HIDN);
  float* h2t  = alloc((size_t)NTXN * HIDN);
  float* emb  = alloc((size_t)NTXN * HIDN);
  float* asrcB = alloc((size_t)NTXN * 4);
  float* adstB = alloc((size_t)NTXN * 4);
  float* mB    = alloc((size_t)NTXN * 4);
  float* sB    = alloc((size_t)NTXN * 4);
  float* aE    = alloc((size_t)NEDGE * 4);
  float* score = alloc(16);
  float* wsem  = alloc(16);
  float* qh   = alloc((size_t)NTXN * HIDN);
  float* ctx  = alloc((size_t)NTXN * HIDN);
  float* ctxo = alloc((size_t)NTXN * HIDN);
  float* seqc = alloc((size_t)TCHROWS * HIDN);
  float* khc  = alloc((size_t)TCHROWS * HIDN);
  float* vhc  = alloc((size_t)TCHROWS * HIDN);
  float* cat  = alloc((size_t)NTXN * 256);
  float* gate = alloc((size_t)NTXN * HIDN);
  float* z1   = alloc((size_t)NTXN * 256);
  float* z2   = alloc((size_t)NTXN * 64);

  auto mkconv = [&](int base) {
    ConvP c;
    for (int i = 0; i < 4; ++i) { c.adst[i] = PF(base + i); c.asrc[i] = PF(base + 4 + i); }
    c.kb = PF(base + 8);  c.kw = PF(base + 9);
    c.pbM = PF(base + 10); c.pwM = PF(base + 11);
    c.pbT = PF(base + 12); c.pwT = PF(base + 13);
    c.pbU = PF(base + 14); c.pwU = PF(base + 15);
    c.q = PF(base + 16);
    return c;
  };
  const ConvP cp1 = mkconv(C1BASE), cp2 = mkconv(C2BASE);
  const int *eiUT = PI(IN_EI_UT), *eiTU = PI(IN_EI_TU);
  const int *eiMT = PI(IN_EI_MT), *eiTM = PI(IN_EI_TM);

  auto gemm = [&](const float* A, const float* W, const float* bias, float* C,
                  int M, int K, int N, int act) {
    dim3 g(M / 16, (N + 127) / 128);
    gemm_bias_act_k<<<g, 256, 0, stream>>>(A, W, bias, C, M, K, N, act);
  };
  auto fill = [&](float* p, float v, long n) {
    fillf_k<<<cdiv(n, 256), 256, 0, stream>>>(p, v, (int)n);
  };

  // one GAT-style edge-type pass: att logits -> segment softmax -> scatter, relu'd out
  auto run_edge = [&](const int* ei, int Ns, int Nd, const float* xps, const float* xpd,
                      const float* attS, const float* attD, float* outb) {
    att_score_k<<<cdiv((long)Ns * 4, 256), 256, 0, stream>>>(xps, attS, asrcB, Ns * 4);
    att_score_k<<<cdiv((long)Nd * 4, 256), 256, 0, stream>>>(xpd, attD, adstB, Nd * 4);
    fill(mB, -1e30f, (long)Nd * 4);
    fill(sB, 0.f, (long)Nd * 4);
    fill(outb, 0.f, (long)Nd * HIDN);
    const int eb = cdiv((long)NEDGE * 4, 256);
    edge_logits_max_k<<<eb, 256, 0, stream>>>(ei, asrcB, adstB, aE, mB, NEDGE);
    edge_exp_sum_k<<<eb, 256, 0, stream>>>(ei, aE, mB, sB, NEDGE);
    edge_norm_k<<<eb, 256, 0, stream>>>(ei, aE, sB, NEDGE);
    edge_scatter_k<<<cdiv((long)NEDGE * HIDN, 256), 256, 0, stream>>>(ei, xps, aE, outb, NEDGE);
    relu_k<<<cdiv((long)Nd * HIDN, 256), 256, 0, stream>>>(outb, Nd * HIDN);
  };

  // full HANConv layer; rev=false skips user/merchant destinations (conv2 dead code)
  auto run_conv = [&](const float* ht, const float* hu, const float* hm, const ConvP& cp,
                      bool rev, float* resTxn) {
    gemm(ht, cp.pwT, cp.pbT, xpt, NTXN, HIDN, HIDN, 0);
    gemm(hu, cp.pwU, cp.pbU, xpu, NUSR, HIDN, HIDN, 0);
    gemm(hm, cp.pwM, cp.pbM, xpm, NMER, HIDN, HIDN, 0);
    run_edge(eiUT, NUSR, NTXN, xpu, xpt, cp.asrc[0], cp.adst[0], ot0);  // pays
    run_edge(eiMT, NMER, NTXN, xpm, xpt, cp.asrc[1], cp.adst[1], ot1);  // receives
    if (rev) {
      run_edge(eiTU, NTXN, NUSR, xpt, xpu, cp.asrc[2], cp.adst[2], ou); // rev_pays
      run_edge(eiTM, NTXN, NMER, xpt, xpm, cp.asrc[3], cp.adst[3], om); // rev_receives
    }
    // semantic attention over txn's 2 metapaths (user/merchant have 1 => weight 1)
    fill(score, 0.f, 2);
    const float invN = 1.f / (float)NTXN;
    semantic_score_k<<<dim3(NTXN / 16), 256, 0, stream>>>(ot0, cp.kw, cp.kb, cp.q, score + 0, invN);
    semantic_score_k<<<dim3(NTXN / 16), 256, 0, stream>>>(ot1, cp.kw, cp.kb, cp.q, score + 1, invN);
    softmax2_k<<<1, 32, 0, stream>>>(score, wsem);
    combine2_k<<<cdiv((long)NTXN * HIDN, 256), 256, 0, stream>>>(ot0, ot1, wsem, resTxn,
                                                                 NTXN * HIDN);
  };

  // ---- stage 0: per-type input projection + BN(eval) + relu ----
  gemm_projbn_k<<<dim3(NTXN / 16), 256, 0, stream>>>(PF(IN_XTXN), PF(PRJBASE + 7),
      PF(PRJBASE + 4), PF(PRJBASE + 6), PF(PRJBASE + 5), h0t, NTXN, 64, HIDN);
  gemm_projbn_k<<<dim3(NUSR / 16), 256, 0, stream>>>(PF(IN_XUSR), PF(PRJBASE + 11),
      PF(PRJBASE + 8), PF(PRJBASE + 10), PF(PRJBASE + 9), h0u, NUSR, 32, HIDN);
  gemm_projbn_k<<<dim3(NMER / 16), 256, 0, stream>>>(PF(IN_XMER), PF(PRJBASE + 3),
      PF(PRJBASE + 0), PF(PRJBASE + 2), PF(PRJBASE + 1), h0m, NMER, 32, HIDN);

  // ---- conv1 + LN1 (relu'd input) ----
  run_conv(h0t, h0u, h0m, cp1, true, rest);
  layernorm128_k<<<cdiv(NTXN, 8), 256, 0, stream>>>(rest, PF(IN_LN1G), PF(IN_LN1B), h1t, NTXN, 1);
  layernorm128_k<<<cdiv(NUSR, 8), 256, 0, stream>>>(ou,   PF(IN_LN1G), PF(IN_LN1B), h1u, NUSR, 1);
  layernorm128_k<<<cdiv(NMER, 8), 256, 0, stream>>>(om,   PF(IN_LN1G), PF(IN_LN1B), h1m, NMER, 1);

  // ---- conv2 (txn dst only) + LN2, residual ----
  run_conv(h1t, h1u, h1m, cp2, false, rest);
  layernorm128_k<<<cdiv(NTXN, 8), 256, 0, stream>>>(rest, PF(IN_LN2G), PF(IN_LN2B), h2t, NTXN, 0);
  add_k<<<cdiv((long)NTXN * HIDN, 256), 256, 0, stream>>>(h2t, h1t, emb, NTXN * HIDN);

  // ---- temporal branch (chunked to bound scratch) ----
  gemm(emb, PF(IN_WQ), PF(IN_BQ), qh, NTXN, HIDN, HIDN, 0);
  const float* seqIn = PF(IN_SEQ);
  const float* dtIn  = PF(IN_DT);
  for (int ch = 0; ch < NTXN / TCH; ++ch) {
    const float* Aseq = seqIn + (size_t)ch * TCHROWS * HIDN;
    const float* dtc  = dtIn + (size_t)ch * TCHROWS;
    gemm_tproj_k<<<dim3(TCHROWS / 16), 256, 0, stream>>>(Aseq, PF(IN_TPROJW), PF(IN_TPROJB),
        dtc, PF(IN_TIMEW), PF(IN_TIMEB), seqc, TCHROWS, HIDN, HIDN);
    gemm(seqc, PF(IN_WK), PF(IN_BK), khc, TCHROWS, HIDN, HIDN, 0);
    gemm(seqc, PF(IN_WV), PF(IN_BV), vhc, TCHROWS, HIDN, HIDN, 0);
    mha10_k<<<TCH / 8, 256, 0, stream>>>(qh + (size_t)ch * TCH * HIDN, khc, vhc,
                                         ctx + (size_t)ch * TCH * HIDN, TCH);
  }
  gemm(ctx, PF(IN_WO), PF(IN_BO), ctxo, NTXN, HIDN, HIDN, 0);

  // ---- gated fusion + MLP head ----
  build_cat_k<<<cdiv((long)NTXN * HIDN, 256), 256, 0, stream>>>(emb, ctxo, cat, NTXN * HIDN);
  gemm(cat, PF(IN_GATE_W), PF(IN_GATE_B), gate, NTXN, 256, HIDN, 2);
  apply_gate_k<<<cdiv((long)NTXN * HIDN, 256), 256, 0, stream>>>(cat, gate, NTXN * HIDN);
  gemm(cat, PF(IN_H1W), PF(IN_H1B), z1, NTXN, 256, 256, 1);
  gemm(z1, PF(IN_H2W), PF(IN_H2B), z2, NTXN, 256, 64, 1);
  head_out_k<<<cdiv(NTXN, 8), 256, 0, stream>>>(z2, PF(IN_H3W), PF(IN_H3B),
                                                (float*)d_out, NTXN);
}